// Encoder_91319594647570
// MI455X (gfx1250) — compile-verified
//
#include <hip/hip_runtime.h>
#include <hip/hip_bf16.h>

#define N_NODES 100000
#define N_EDGES 1600000
#define HID 64

typedef float v2f __attribute__((ext_vector_type(2)));
typedef float v8f __attribute__((ext_vector_type(8)));

// ---------------- Kernel 1: init degree accumulator (self loop = 1) --------
__global__ void init_deg_kernel(float* __restrict__ deg) {
    int i = blockIdx.x * blockDim.x + threadIdx.x;
    if (i < N_NODES) deg[i] = 1.0f;
}

// ---------------- Kernel 2: accumulate in-degree over edges ---------------
__global__ void edge_deg_kernel(const long long* __restrict__ ei, float* __restrict__ deg) {
    int e = blockIdx.x * blockDim.x + threadIdx.x;
    if (e >= N_EDGES) return;
    long long d = ei[(size_t)N_EDGES + e];
    atomicAdd(&deg[d], 1.0f);
}

// ------- Kernel 3: dis = rsqrt(deg) (in place) ; xa = dis^2 * x (self) ----
__global__ void node_dis_xa_kernel(float* __restrict__ dis, const float* __restrict__ x,
                                   float* __restrict__ xa) {
    int i = blockIdx.x * blockDim.x + threadIdx.x;
    if (i >= N_NODES) return;
    float d = dis[i];                 // currently holds deg
    float r = rsqrtf(d);
    dis[i] = r;
    float r2 = r * r;
    xa[i * 2 + 0] = r2 * x[i * 2 + 0];
    xa[i * 2 + 1] = r2 * x[i * 2 + 1];
}

// ---------------- Kernel 4: edge-aggregate x (2 channels) -----------------
__global__ void edge_xa_kernel(const long long* __restrict__ ei, const float* __restrict__ dis,
                               const float* __restrict__ x, float* __restrict__ xa) {
    int e = blockIdx.x * blockDim.x + threadIdx.x;
    if (e >= N_EDGES) return;
    long long s = ei[e];
    long long d = ei[(size_t)N_EDGES + e];
    float norm = dis[s] * dis[d];
    atomicAdd(&xa[d * 2 + 0], norm * x[s * 2 + 0]);
    atomicAdd(&xa[d * 2 + 1], norm * x[s * 2 + 1]);
}

// ----- Kernel 5: h = relu(xa @ W1 + b1); seed g with self-loop term -------
__global__ void node_h_kernel(const float* __restrict__ xa, const float* __restrict__ dis,
                              const float* __restrict__ W1, const float* __restrict__ b1,
                              float* __restrict__ h, float* __restrict__ gacc) {
    int t = blockIdx.x * blockDim.x + threadIdx.x;
    if (t >= N_NODES * HID) return;
    int i = t / HID;
    int j = t - i * HID;
    float v = xa[i * 2 + 0] * W1[j] + xa[i * 2 + 1] * W1[HID + j] + b1[j];
    v = v > 0.0f ? v : 0.0f;
    h[t] = v;
    float r = dis[i];
    gacc[t] = r * r * v;              // self-loop contribution to aggregation
}

// -------- Kernel 6: wave-per-edge aggregation of h (64 channels) ----------
__global__ void edge_agg_kernel(const long long* __restrict__ ei, const float* __restrict__ dis,
                                const float* __restrict__ h, float* __restrict__ gacc) {
    int e = (blockIdx.x * blockDim.x + threadIdx.x) >> 5;
    int lane = threadIdx.x & 31;
    if (e >= N_EDGES) return;
    long long s = ei[e];
    long long d = ei[(size_t)N_EDGES + e];
    float norm = dis[s] * dis[d];
    float2 v = ((const float2*)(h + (size_t)s * HID))[lane];
    atomicAdd(&gacc[(size_t)d * HID + lane * 2 + 0], norm * v.x);
    atomicAdd(&gacc[(size_t)d * HID + lane * 2 + 1], norm * v.y);
}

// -------- Kernel 7: WMMA GEMM  out[100000,16]x2 = g[100000,64] @ W[64,16] -
// One wave per 16x16 output tile. V_WMMA_F32_16X16X4_F32, 16 steps over K=64.
__global__ void __launch_bounds__(256) gemm_heads_kernel(
    const float* __restrict__ gacc,
    const float* __restrict__ W_mu, const float* __restrict__ b_mu,
    const float* __restrict__ W_ls, const float* __restrict__ b_ls,
    float* __restrict__ out) {
    const int nMTiles = N_NODES / 16;                 // 6250, exact
    int wave = blockIdx.x * (blockDim.x >> 5) + (threadIdx.x >> 5);
    if (wave >= nMTiles * 2) return;                  // uniform per wave (EXEC all-1s kept)
    int lane = threadIdx.x & 31;

    int nt = wave & 1;                                // 0 -> mu head, 1 -> logstd head
    int mt = wave >> 1;
    const float* W    = nt ? W_ls : W_mu;             // [64,16] row-major
    const float* bias = nt ? b_ls : b_mu;
    float* outp = out + (size_t)nt * (size_t)N_NODES * 16;

    int col  = lane & 15;
    int kh   = (lane >> 4) * 2;                       // K sub-offset: 0 or 2
    const float* grow = gacc + (size_t)(mt * 16 + col) * HID;  // A row = lane%16
    float bcol = bias[col];

    v8f c = {0.f, 0.f, 0.f, 0.f, 0.f, 0.f, 0.f, 0.f};
#pragma unroll
    for (int kk = 0; kk < HID; kk += 4) {
        v2f a, b;
        a.x = grow[kk + kh];
        a.y = grow[kk + kh + 1];
        b.x = W[(kk + kh) * 16 + col];
        b.y = W[(kk + kh + 1) * 16 + col];
        c = __builtin_amdgcn_wmma_f32_16x16x4_f32(
            /*neg_a=*/false, a, /*neg_b=*/false, b,
            /*c_mod=*/(short)0, c, /*reuse_a=*/false, /*reuse_b=*/false);
    }

    int mbase = mt * 16 + (lane >> 4) * 8;            // D layout: VGPR v -> M = v (+8 hi half)
#pragma unroll
    for (int v = 0; v < 8; ++v) {
        outp[(size_t)(mbase + v) * 16 + col] = c[v] + bcol;
    }
}

extern "C" void kernel_launch(void* const* d_in, const int* in_sizes, int n_in,
                              void* d_out, int out_size, void* d_ws, size_t ws_size,
                              hipStream_t stream) {
    const float*     x    = (const float*)d_in[0];
    const long long* ei   = (const long long*)d_in[1];   // int64 [2, E]
    const float*     W1   = (const float*)d_in[2];       // [2, 64]
    const float*     b1   = (const float*)d_in[3];       // [64]
    const float*     W_mu = (const float*)d_in[4];       // [64, 16]
    const float*     b_mu = (const float*)d_in[5];       // [16]
    const float*     W_ls = (const float*)d_in[6];       // [64, 16]
    const float*     b_ls = (const float*)d_in[7];       // [16]
    float* out = (float*)d_out;                          // [mu | logstd], each [100000,16]

    // Workspace layout (floats): dis[N] | xa[2N] | h[64N] | g[64N]  = 131N = 52.4 MB
    float* dis  = (float*)d_ws;
    float* xa   = dis + N_NODES;
    float* h    = xa + 2 * N_NODES;
    float* gacc = h + (size_t)HID * N_NODES;

    const int T = 256;
    int nodeBlocks = (N_NODES + T - 1) / T;
    int edgeBlocks = (N_EDGES + T - 1) / T;
    int hBlocks    = (N_NODES * HID + T - 1) / T;
    int aggBlocks  = (int)(((size_t)N_EDGES * 32 + T - 1) / T);
    int gemmWaves  = (N_NODES / 16) * 2;                 // 12500
    int gemmBlocks = (gemmWaves + (T / 32) - 1) / (T / 32);

    init_deg_kernel<<<nodeBlocks, T, 0, stream>>>(dis);
    edge_deg_kernel<<<edgeBlocks, T, 0, stream>>>(ei, dis);
    node_dis_xa_kernel<<<nodeBlocks, T, 0, stream>>>(dis, x, xa);
    edge_xa_kernel<<<edgeBlocks, T, 0, stream>>>(ei, dis, x, xa);
    node_h_kernel<<<hBlocks, T, 0, stream>>>(xa, dis, W1, b1, h, gacc);
    edge_agg_kernel<<<aggBlocks, T, 0, stream>>>(ei, dis, h, gacc);
    gemm_heads_kernel<<<gemmBlocks, T, 0, stream>>>(gacc, W_mu, b_mu, W_ls, b_ls, out);
}